// DynamicGaussianBlur_4896262718029
// MI455X (gfx1250) — compile-verified
//
#include <hip/hip_runtime.h>
#include <math.h>

// ---------------------------------------------------------------------------
// Dynamic per-batch separable 3D Gaussian blur, CDNA5 (gfx1250, wave32).
//
// 3 bandwidth-bound passes (D, H, W), each a band-matrix GEMM per tile:
//   out(16x16) = G(16x28) * In(28x16)  via 7x V_WMMA_F32_16X16X4_F32.
// - A (Gaussian band) lives in 7 float2 register chunks per wave.
// - B uses a sliding register window: per tile only 16 new rows (4 chunks)
//   are loaded; the other 12 rows are register-shifted (Bv[0..2]=Bv[4..6]).
// - Stride S is a template parameter: no runtime multiplies in addressing.
// - Boundary handling is branch-free (clamp + cndmask) and only exists in
//   the first/last tile after full unroll.
// ---------------------------------------------------------------------------

typedef __attribute__((ext_vector_type(2))) float v2f;
typedef __attribute__((ext_vector_type(8))) float v8f;

#define AXIS_LEN    160
#define RAD         6
#define WIN         13
#define BATCH_ELEMS (160 * 160 * 160 * 2)    // 8,192,000 elems per batch sample
#define TOTAL_ELEMS (4 * BATCH_ELEMS)        // 32,768,000
#define TOTAL_LINES (TOTAL_ELEMS / AXIS_LEN) // 204,800 per pass
#define NGROUPS     (TOTAL_LINES / 16)       // 12,800 wave tiles per pass

// --- 1) Precompute normalized Gaussian taps (exact reference formula) -------
__global__ void gauss_wts_kernel(const float* __restrict__ sigma,
                                 float* __restrict__ wts) {
    int id = threadIdx.x;            // one thread per (batch, axis)
    if (id >= 12) return;
    int b = id / 3, axis = id % 3;
    float s = sigma[b * 3 + axis];
    float denom = 2.0f * s * s + 1e-7f;
    float lg = logf(sqrtf(2.0f * (float)M_PI) * s + 1e-7f);
    float g[WIN];
    float sum = 0.0f;
    #pragma unroll
    for (int t = 0; t < WIN; ++t) {
        float loc = (float)t - (float)RAD;
        g[t] = expf(-(loc * loc) / denom - lg);
        sum += g[t];
    }
    float inv = 1.0f / sum;
    #pragma unroll
    for (int t = 0; t < WIN; ++t)
        wts[(b * 3 + axis) * WIN + t] = g[t] * inv;
}

// Branch-free boundary loads: clamp index so the load is always legal, then
// zero out-of-range values with a select (v_cndmask), never an EXEC branch.
template <int S>
__device__ __forceinline__ float ld_clamp_lo(const float* __restrict__ line, int c) {
    int cc = c < 0 ? 0 : c;                 // only low side can be OOB here
    float v = line[cc * S];
    return (c < 0) ? 0.0f : v;
}
template <int S>
__device__ __forceinline__ float ld_clamp_hi(const float* __restrict__ line, int c) {
    int cc = c > (AXIS_LEN - 1) ? (AXIS_LEN - 1) : c;   // only high side OOB
    float v = line[cc * S];
    return (c >= AXIS_LEN) ? 0.0f : v;
}

// --- 2) One separable pass along an axis of compile-time stride S -----------
template <int S>
__global__ __launch_bounds__(128) void blur_pass_kernel(
        const float* __restrict__ src, float* __restrict__ dst,
        const float* __restrict__ wts, int axis) {
    const int lane  = threadIdx.x & 31;
    const int wave  = threadIdx.x >> 5;
    const int group = blockIdx.x * 4 + wave;   // grid sized exactly: no guard

    const int n    = lane & 15;   // B/C/D column = line within group
    const int half = lane >> 4;   // selects K rows {0,1} vs {2,3} per chunk
    // Line id -> element base (16-line groups never straddle batches).
    const int q     = group * 16 + n;
    const int outer = q / S;                       // constant-divisor magic mul
    const int inner = q - outer * S;
    const int base  = outer * (S * AXIS_LEN) + inner;
    const int b     = base / BATCH_ELEMS;
    const float* __restrict__ g = wts + (b * 3 + axis) * WIN;

    // Band matrix A: G[m, j] = g[j - m] for 0 <= j-m < 13, j in [0,28).
    // A 16x4 f32 layout: lane row m = lane&15; v0/v1 hold K = 4kc+2*half+{0,1}.
    const int m = n;
    v2f A[7];
    #pragma unroll
    for (int kc = 0; kc < 7; ++kc) {
        int j0 = 4 * kc + 2 * half;
        int t0 = j0 - m;
        int t1 = t0 + 1;
        A[kc].x = (t0 >= 0 && t0 < WIN) ? g[t0] : 0.0f;
        A[kc].y = (t1 >= 0 && t1 < WIN) ? g[t1] : 0.0f;
    }

    const float* __restrict__ line  = src + base;
    float* __restrict__       oline = dst + base + (8 * half) * S; // store base

    // Prologue: B rows for tile 0, global coords c in [-6, 21] (low clamp).
    v2f Bv[7];
    #pragma unroll
    for (int kc = 0; kc < 7; ++kc) {
        int c0 = 4 * kc + 2 * half - RAD;
        Bv[kc].x = ld_clamp_lo<S>(line, c0);
        Bv[kc].y = ld_clamp_lo<S>(line, c0 + 1);
    }

    #pragma unroll
    for (int t = 0; t < 10; ++t) {
        const int i0 = 16 * t;

        v8f acc = {};
        #pragma unroll
        for (int kc = 0; kc < 7; ++kc) {
            acc = __builtin_amdgcn_wmma_f32_16x16x4_f32(
                /*neg_a=*/false, A[kc], /*neg_b=*/false, Bv[kc],
                /*c_mod=*/(short)0, acc, /*reuse_a=*/false, /*reuse_b=*/false);
        }

        // Sliding window: next tile's rows 0..11 are current rows 16..27
        // (chunk shift by 4); fetch only the 16 fresh rows (chunks 3..6).
        if (t < 9) {
            Bv[0] = Bv[4];
            Bv[1] = Bv[5];
            Bv[2] = Bv[6];
            #pragma unroll
            for (int kc = 0; kc < 4; ++kc) {
                int c0 = i0 + 22 + 4 * kc + 2 * half;   // fresh rows
                if (t == 8) {   // constant after unroll: only last refill masks
                    Bv[3 + kc].x = ld_clamp_hi<S>(line, c0);
                    Bv[3 + kc].y = ld_clamp_hi<S>(line, c0 + 1);
                } else {
                    Bv[3 + kc].x = line[c0 * S];
                    Bv[3 + kc].y = line[(c0 + 1) * S];
                }
            }
        }

        // D 16x16 f32 layout: VGPR v -> row v + 8*half, column n.
        #pragma unroll
        for (int v = 0; v < 8; ++v)
            oline[(i0 + v) * S] = acc[v];
    }
}

// --- 3) Orchestration --------------------------------------------------------
extern "C" void kernel_launch(void* const* d_in, const int* in_sizes, int n_in,
                              void* d_out, int out_size, void* d_ws, size_t ws_size,
                              hipStream_t stream) {
    const float* image = (const float*)d_in[0];   // [4,160,160,160,2] f32
    const float* sigma = (const float*)d_in[1];   // [4,3] f32
    float* out = (float*)d_out;
    float* wts = (float*)d_ws;                    // 156 floats
    float* tmp = (float*)((char*)d_ws + 4096);    // 131 MB ping-pong

    gauss_wts_kernel<<<1, 32, 0, stream>>>(sigma, wts);

    dim3 block(128);            // 4 wave32 per block
    dim3 grid(NGROUPS / 4);     // 3200 blocks, one wave per 16 lines

    constexpr int S_D = 160 * 160 * 2;  // depth-axis element stride
    constexpr int S_H = 160 * 2;        // height-axis element stride
    constexpr int S_W = 2;              // width-axis element stride (C innermost)

    // image --D--> out --H--> tmp --W--> out
    blur_pass_kernel<S_D><<<grid, block, 0, stream>>>(image, out, wts, 0);
    blur_pass_kernel<S_H><<<grid, block, 0, stream>>>(out,   tmp, wts, 1);
    blur_pass_kernel<S_W><<<grid, block, 0, stream>>>(tmp,   out, wts, 2);
}